// CtcLossLayer_44263932953116
// MI455X (gfx1250) — compile-verified
//
#include <hip/hip_runtime.h>
#include <hip/hip_bf16.h>

typedef float v2f __attribute__((ext_vector_type(2)));
typedef float v8f __attribute__((ext_vector_type(8)));

#define NEG_INF (-1e30f)

// Problem constants (fixed by the reference harness)
#define BB 32
#define TT 512
#define CC 1000
#define LL 64
#define SS 129      // 2*L + 1
#define SP 132      // padded row stride for lp_ext
#define BLANK (CC - 1)

// ---------------------------------------------------------------------------
// Kernel 1: fused log-softmax + extended-label gather.
// One block handles 16 consecutive (b,t) rows (16 | T so b is uniform).
//  - tile staged HBM -> LDS via global_load_async_to_lds_b128 (ASYNCcnt path)
//  - row sums of exp(x-max) via v_wmma_f32_16x16x4_f32, B = ones,
//    2 independent accumulators to break the WMMA D->C dependency chain.
// ---------------------------------------------------------------------------
__global__ __launch_bounds__(256) void ctc_lse_gather(
    const float* __restrict__ logits,   // [B,T,C]
    const int*   __restrict__ labels,   // [B,L]
    float*       __restrict__ lpext)    // [B,T,SP]
{
    __shared__ __align__(16) float sdata[16 * CC];   // 64000 B tile of logits
    __shared__ float smax[16];
    __shared__ float spart[8 * 16];                  // per-wave partial row sums
    __shared__ float slse[16];

    const int tid  = threadIdx.x;
    const int lane = tid & 31;
    const int wave = tid >> 5;
    const int row0 = blockIdx.x * 16;   // global row index = b*T + t
    const int b    = row0 / TT;         // uniform across block

    // ---- stage 16 rows (16000 floats = 4000 x b128) HBM -> LDS, async ----
    {
        const float*   gbase = logits + (size_t)row0 * CC;
        const unsigned sbase = (unsigned)(size_t)(&sdata[0]);  // LDS offset = addr[31:0]
        for (int idx = tid; idx < 4000; idx += 256) {
            const float*   gp = gbase + idx * 4;
            const unsigned la = sbase + (unsigned)idx * 16u;
            asm volatile("global_load_async_to_lds_b128 %0, %1, off"
                         :: "v"(la), "v"(gp) : "memory");
        }
        asm volatile("s_wait_asynccnt 0x0" ::: "memory");
    }
    __syncthreads();

    // ---- per-row max: wave w -> rows 2w (lanes 0-15) and 2w+1 (lanes 16-31) ----
    {
        const int r   = 2 * wave + (lane >> 4);
        const int l16 = lane & 15;
        const float* rowp = sdata + r * CC;
        float m = NEG_INF;
        for (int c = l16; c < CC; c += 16) m = fmaxf(m, rowp[c]);
        for (int off = 1; off < 16; off <<= 1)
            m = fmaxf(m, __shfl_xor(m, off, 32));   // xor<16 stays in half-wave
        if (l16 == 0) smax[r] = m;
    }
    __syncthreads();

    // ---- sum of exp(x - max) via WMMA f32 16x16x4, B = all-ones ----
    // A layout (f32 16x4): lanes 0-15 hold M=lane, K=0,1 in v0,v1;
    //                      lanes 16-31 hold M=lane-16, K=2,3.
    {
        const int M      = lane & 15;
        const int khalf  = (lane >> 4) ? 2 : 0;
        const float rmax = smax[M];
        const float* rowbase = sdata + M * CC;
        const v2f bones = {1.0f, 1.0f};
        v8f c0 = {0.f, 0.f, 0.f, 0.f, 0.f, 0.f, 0.f, 0.f};
        v8f c1 = {0.f, 0.f, 0.f, 0.f, 0.f, 0.f, 0.f, 0.f};

        int ch = wave;                          // chunks ch, ch+8, ... < 250
        for (; ch + 8 < (CC / 4); ch += 16) {   // 2 chunks per iteration
            const int ka = ch * 4 + khalf;
            const int kb = ka + 32;             // (ch+8)*4 + khalf
            const float2 xa = *(const float2*)(rowbase + ka);  // 8B aligned
            const float2 xb = *(const float2*)(rowbase + kb);
            v2f a0, a1;
            a0[0] = __expf(xa.x - rmax);
            a0[1] = __expf(xa.y - rmax);
            a1[0] = __expf(xb.x - rmax);
            a1[1] = __expf(xb.y - rmax);
            c0 = __builtin_amdgcn_wmma_f32_16x16x4_f32(
                     false, a0, false, bones, (short)0, c0, false, false);
            c1 = __builtin_amdgcn_wmma_f32_16x16x4_f32(
                     false, a1, false, bones, (short)0, c1, false, false);
        }
        if (ch < (CC / 4)) {                    // odd-count tail
            const int ka = ch * 4 + khalf;
            const float2 xa = *(const float2*)(rowbase + ka);
            v2f a0;
            a0[0] = __expf(xa.x - rmax);
            a0[1] = __expf(xa.y - rmax);
            c0 = __builtin_amdgcn_wmma_f32_16x16x4_f32(
                     false, a0, false, bones, (short)0, c0, false, false);
        }
        c0 = c0 + c1;

        // D layout: lane 0 -> D[M=j][N=0] in c[j]; lane 16 -> D[M=8+j][N=0]
        if (lane == 0)
            for (int j = 0; j < 8; ++j) spart[wave * 16 + j] = c0[j];
        if (lane == 16)
            for (int j = 0; j < 8; ++j) spart[wave * 16 + 8 + j] = c0[j];
    }
    __syncthreads();

    if (tid < 16) {
        float tot = 0.f;
        for (int w = 0; w < 8; ++w) tot += spart[w * 16 + tid];
        slse[tid] = __logf(tot);
    }
    __syncthreads();

    // ---- gather extended-label emission log-probs: 16 rows x 129 states ----
    const int* lab = labels + b * LL;
    for (int p = tid; p < 16 * SS; p += 256) {
        const int m = p / SS;
        const int s = p - m * SS;
        const int cls = (s & 1) ? lab[s >> 1] : BLANK;
        const float v = sdata[m * CC + cls] - smax[m] - slse[m];
        lpext[(size_t)(row0 + m) * SP + s] = v;
    }
}

// ---------------------------------------------------------------------------
// Kernel 2: CTC forward recursion. One block per batch element.
// alpha double-buffered in LDS; one barrier per time step; lp row
// prefetched one step ahead (L2-resident workspace).
// ---------------------------------------------------------------------------
__global__ __launch_bounds__(160) void ctc_forward(
    const float* __restrict__ lpext,        // [B,T,SP]
    const int*   __restrict__ labels,       // [B,L]
    const int*   __restrict__ label_length, // [B]
    const int*   __restrict__ logit_length, // [B]
    float*       __restrict__ loss)         // [B]
{
    __shared__ float buf[2][SP];
    __shared__ int   ext[SP];

    const int b = blockIdx.x;
    const int s = threadIdx.x;
    const size_t base = (size_t)b * TT * SP;

    if (s < SS)
        ext[s] = (s & 1) ? labels[b * LL + (s >> 1)] : BLANK;
    __syncthreads();

    bool skip_ok = false;
    if (s < SS) {
        skip_ok = (s >= 2) && (ext[s] != ext[s - 2]);
        float a0 = NEG_INF;
        if (s == 0) a0 = lpext[base + 0];
        if (s == 1) a0 = lpext[base + 1];
        buf[0][s] = a0;
    }
    __syncthreads();

    const int tl   = logit_length[b];          // uniform across block
    const int tend = (tl < TT) ? tl : TT;      // frozen after tl -> stop early
    int cur = 0;

    float lp_next = 0.f;
    if (s < SS && tend > 1) lp_next = lpext[base + (size_t)SP + s];

    for (int t = 1; t < tend; ++t) {
        const float lp = lp_next;
        if (s < SS && (t + 1) < tend)
            lp_next = lpext[base + (size_t)(t + 1) * SP + s];

        float nv = NEG_INF;
        if (s < SS) {
            const float a1 = buf[cur][s];
            const float a2 = (s >= 1) ? buf[cur][s - 1] : NEG_INF;
            const float a3 = skip_ok ? buf[cur][s - 2] : NEG_INF;
            const float m  = fmaxf(a1, fmaxf(a2, a3));
            const float sm = __expf(a1 - m) + __expf(a2 - m) + __expf(a3 - m);
            nv = m + __logf(sm) + lp;
        }
        if (s < SS) buf[cur ^ 1][s] = nv;
        cur ^= 1;
        __syncthreads();
    }

    if (s == 0) {
        const int ll = label_length[b];
        const float ab = buf[cur][2 * ll];
        const float al = (ll > 0) ? buf[cur][2 * ll - 1] : NEG_INF;
        const float m  = fmaxf(ab, al);
        loss[b] = -(m + __logf(__expf(ab - m) + __expf(al - m)));
    }
}

// ---------------------------------------------------------------------------
// Kernel 3: mean over batch (single wave32).
// ---------------------------------------------------------------------------
__global__ __launch_bounds__(32) void ctc_mean(
    const float* __restrict__ loss, float* __restrict__ out)
{
    float v = loss[threadIdx.x];
    for (int off = 16; off >= 1; off >>= 1) v += __shfl_xor(v, off, 32);
    if (threadIdx.x == 0) out[0] = v * (1.0f / BB);
}

// ---------------------------------------------------------------------------
extern "C" void kernel_launch(void* const* d_in, const int* in_sizes, int n_in,
                              void* d_out, int out_size, void* d_ws, size_t ws_size,
                              hipStream_t stream) {
    (void)in_sizes; (void)n_in; (void)out_size; (void)ws_size;
    const int*   labels       = (const int*)d_in[0];
    const float* logits       = (const float*)d_in[1];
    const int*   label_length = (const int*)d_in[2];
    const int*   logit_length = (const int*)d_in[3];
    float*       out          = (float*)d_out;

    float* lpext = (float*)d_ws;                       // B*T*SP floats (~8.65 MB)
    float* loss  = lpext + (size_t)BB * TT * SP;       // B floats

    ctc_lse_gather<<<(BB * TT) / 16, 256, 0, stream>>>(logits, labels, lpext);
    ctc_forward<<<BB, 160, 0, stream>>>(lpext, labels, label_length, logit_length, loss);
    ctc_mean<<<1, 32, 0, stream>>>(loss, out);
}